// get_loss_20023137534169
// MI455X (gfx1250) — compile-verified
//
#include <hip/hip_runtime.h>
#include <math.h>

// Chamfer distance via V_WMMA_F32_16X16X4_F32 with the norm folded into K=4:
//   A row m = (-2x, -2y, -2z, 1)   (query, 16x4)
//   B col n = ( x,   y,   z, |t|^2) (reference, 4x16)
//   D = A x B + 0  =  |t_n|^2 - 2 p_m . t_n          (C = inline 0)
//   running v_min3 across tile pairs; add |p_m|^2 after the final row-min.

typedef float v2f __attribute__((ext_vector_type(2)));
typedef float v8f __attribute__((ext_vector_type(8)));

#define QB 4            // 4 row-blocks of 16 => 64 query points per wave
#define BATCH 4
#define NPTS 8192

__device__ __forceinline__ float min3_num(float a, float b, float c) {
    float d;
    asm("v_min3_num_f32 %0, %1, %2, %3" : "=v"(d) : "v"(a), "v"(b), "v"(c));
    return d;
}
__device__ __forceinline__ float min_num(float a, float b) {
    float d;
    asm("v_min_num_f32 %0, %1, %2" : "=v"(d) : "v"(a), "v"(b));
    return d;
}

__global__ __launch_bounds__(32) void chamfer_min_dir(
    const float* __restrict__ pred, const float* __restrict__ target,
    float* __restrict__ ws)
{
    const int lane = threadIdx.x;   // wave32
    const int col  = lane & 15;     // column within 16-wide tile
    const int hi   = lane >> 4;     // 0: K=0,1 / rows 0..7 ; 1: K=2,3 / rows 8..15

    const int bpb = NPTS / (16 * QB);             // qblocks per batch
    const int dir = blockIdx.x / (BATCH * bpb);   // 0: pred->target, 1: target->pred
    const int rem = blockIdx.x % (BATCH * bpb);
    const int b    = rem / bpb;
    const int qblk = rem % bpb;

    const float* qb = (dir ? target : pred) + (size_t)b * NPTS * 3;
    const float* rb = (dir ? pred : target) + (size_t)b * NPTS * 3;
    float* outmin = ws + (size_t)dir * BATCH * NPTS + (size_t)b * NPTS;

    v2f   a[QB];
    float qn[QB];
    v8f   runmin[QB];

#pragma unroll
    for (int j = 0; j < QB; ++j) {
        const int row = qblk * (16 * QB) + j * 16 + col;
        const float x = qb[row * 3 + 0];
        const float y = qb[row * 3 + 1];
        const float z = qb[row * 3 + 2];
        qn[j]  = x * x + y * y + z * z;
        // A (16x4 f32): lanes 0-15 -> K=0,1 = (-2x,-2y) ; lanes 16-31 -> K=2,3 = (-2z, 1)
        a[j].x = hi ? (-2.0f * z) : (-2.0f * x);
        a[j].y = hi ? 1.0f        : (-2.0f * y);
#pragma unroll
        for (int e = 0; e < 8; ++e) runmin[j][e] = 3.0e38f;
    }

    const v8f czero = {};                // C = inline 0 in the WMMA encoding

    const int npair = (NPTS / 16) / 2;   // tile pairs; pair p covers cols p*32 .. p*32+31

    // software pipeline: preload pair 0
    float ax0, ay0, az0, ax1, ay1, az1;
    {
        const float* p0 = rb + (size_t)col * 3;
        ax0 = p0[0]; ay0 = p0[1]; az0 = p0[2];
        const float* p1 = rb + (size_t)(16 + col) * 3;
        ax1 = p1[0]; ay1 = p1[1]; az1 = p1[2];
    }

    for (int p = 0; p < npair; ++p) {
        // prefetch next pair (clamped) so loads overlap compute
        const int pnx = (p + 1 < npair) ? (p + 1) : p;
        const float* n0 = rb + (size_t)(pnx * 32 + col) * 3;
        const float nx0 = n0[0], ny0 = n0[1], nz0 = n0[2];
        const float* n1 = rb + (size_t)(pnx * 32 + 16 + col) * 3;
        const float nx1 = n1[0], ny1 = n1[1], nz1 = n1[2];

        const float tn0 = ax0 * ax0 + ay0 * ay0 + az0 * az0;
        const float tn1 = ax1 * ax1 + ay1 * ay1 + az1 * az1;
        // B (4x16 f32): lanes 0-15 -> K=0,1 = (x,y) ; lanes 16-31 -> K=2,3 = (z, |t|^2)
        v2f bm0, bm1;
        bm0.x = hi ? az0 : ax0;  bm0.y = hi ? tn0 : ay0;
        bm1.x = hi ? az1 : ax1;  bm1.y = hi ? tn1 : ay1;

#pragma unroll
        for (int j = 0; j < QB; ++j) {
            v8f d0 = __builtin_amdgcn_wmma_f32_16x16x4_f32(
                         false, a[j], false, bm0, (short)0, czero, false, false);
            v8f d1 = __builtin_amdgcn_wmma_f32_16x16x4_f32(
                         false, a[j], false, bm1, (short)0, czero, false, false);
#pragma unroll
            for (int e = 0; e < 8; ++e)
                runmin[j][e] = min3_num(runmin[j][e], d0[e], d1[e]);
        }

        ax0 = nx0; ay0 = ny0; az0 = nz0;
        ax1 = nx1; ay1 = ny1; az1 = nz1;
    }

    // Row-min: reduce over the 16 column lanes within each half-wave.
#pragma unroll
    for (int j = 0; j < QB; ++j) {
#pragma unroll
        for (int e = 0; e < 8; ++e) {
            float v = runmin[j][e];
            v = min_num(v, __shfl_xor(v, 1));
            v = min_num(v, __shfl_xor(v, 2));
            v = min_num(v, __shfl_xor(v, 4));
            v = min_num(v, __shfl_xor(v, 8));
            // add |q_row|^2 for this register's row (row = e + 8*hi, norm in lane e+8*hi)
            const float qnorm = __shfl(qn[j], e + 8 * hi);
            if (col == 0) {
                const int row = qblk * (16 * QB) + j * 16 + e + 8 * hi;
                outmin[row] = v + qnorm;
            }
        }
    }
}

__global__ __launch_bounds__(256) void chamfer_finalize(
    const float* __restrict__ mins, int n,
    const float* __restrict__ bpp, const float* __restrict__ lamda,
    float* __restrict__ out, float scale)
{
    __shared__ float red[256];
    const int tid = threadIdx.x;
    float s = 0.0f;
    for (int i = tid; i < n; i += 256) s += mins[i];
    red[tid] = s;
    __syncthreads();
    for (int off = 128; off > 0; off >>= 1) {
        if (tid < off) red[tid] += red[tid + off];
        __syncthreads();
    }
    if (tid == 0) out[0] = red[0] * scale + lamda[0] * bpp[0];
}

extern "C" void kernel_launch(void* const* d_in, const int* in_sizes, int n_in,
                              void* d_out, int out_size, void* d_ws, size_t ws_size,
                              hipStream_t stream)
{
    const float* pred   = (const float*)d_in[0];
    const float* target = (const float*)d_in[1];
    const float* bpp    = (const float*)d_in[2];
    const float* lamda  = (const float*)d_in[3];
    float* out = (float*)d_out;

    float* mins = (float*)d_ws;   // [2][BATCH][NPTS]: dir0 = pred->target, dir1 = target->pred

    const int total_blocks = 2 * BATCH * (NPTS / (16 * QB));   // 1024 single-wave blocks
    chamfer_min_dir<<<dim3(total_blocks), dim3(32), 0, stream>>>(pred, target, mins);

    // result = (sum over both dirs) / (BATCH*NPTS) + lamda*bpp   (N == M)
    chamfer_finalize<<<1, 256, 0, stream>>>(mins, 2 * BATCH * NPTS, bpp, lamda, out,
                                            1.0f / (float)(BATCH * NPTS));
}